// TensorQuantizer_60095182406150
// MI455X (gfx1250) — compile-verified
//
#include <hip/hip_runtime.h>
#include <stdint.h>

// Tensor shape from the reference: (OC=1024, IC=1024, 3, 3) fp32
#define OC 1024
#define IC 1024
#define KK 9
#define NROWS (OC * IC)        // 1,048,576 kernel rows
#define NELEM (NROWS * KK)     // 9,437,184 elements

typedef __attribute__((ext_vector_type(2))) float v2f;
typedef __attribute__((ext_vector_type(8))) float v8f;

__device__ __forceinline__ float waveReduceSum(float v) {
  #pragma unroll
  for (int off = 16; off > 0; off >>= 1) v += __shfl_xor(v, off, 32);
  return v;
}
__device__ __forceinline__ float waveReduceMax(float v) {
  #pragma unroll
  for (int off = 16; off > 0; off >>= 1) v = fmaxf(v, __shfl_xor(v, off, 32));
  return v;
}

// ---------------------------------------------------------------- init ws
__global__ void k_init(unsigned* w) { w[0] = 0u; }

// ---------------------------------------------------------------- global max|x|
__global__ __launch_bounds__(256) void k_maxabs(const float* __restrict__ x,
                                                unsigned* __restrict__ maxbits) {
  __shared__ float smax[8];
  float m = 0.f;
  const float4* x4 = (const float4*)x;
  const int n4 = NELEM / 4;
  for (int i = blockIdx.x * blockDim.x + threadIdx.x; i < n4;
       i += gridDim.x * blockDim.x) {
    float4 v = x4[i];
    m = fmaxf(m, fmaxf(fmaxf(fabsf(v.x), fabsf(v.y)),
                       fmaxf(fabsf(v.z), fabsf(v.w))));
  }
  m = waveReduceMax(m);
  int lane = threadIdx.x & 31, wid = threadIdx.x >> 5;
  if (lane == 0) smax[wid] = m;
  __syncthreads();
  if (wid == 0) {
    m = (lane < 8) ? smax[lane] : 0.f;
    m = waveReduceMax(m);
    // nonnegative floats compare identically as unsigned bit patterns
    if (lane == 0) atomicMax(maxbits, __float_as_uint(m));
  }
}

// ---------------------------------------------------------------- stage 1 (squant-k)
// One thread per 9-element kernel row; one block handles 256 rows = 9216 bytes,
// staged via CDNA5 async global<->LDS DMA.
__global__ __launch_bounds__(256) void k_stage1(const float* __restrict__ x,
                                                float* __restrict__ out,
                                                const unsigned* __restrict__ maxbits,
                                                float* __restrict__ e_row,
                                                float* __restrict__ b_prio,
                                                int* __restrict__ b_meta) {
  __shared__ float tile[256 * KK];                 // 9216 bytes
  const int t = threadIdx.x;
  const unsigned blockBytes = 256u * KK * 4u;      // 9216, 16B aligned
  const unsigned ldsbase = (unsigned)(uintptr_t)(&tile[0]);  // low 32 bits = LDS offset
  const uint64_t gbase =
      (uint64_t)(uintptr_t)x + (uint64_t)blockIdx.x * blockBytes;

  // async stage-in: 576 chunks of 16B
  #pragma unroll
  for (int i = 0; i < 3; ++i) {
    int idx = t + 256 * i;
    if (idx < 576) {
      unsigned off = (unsigned)idx * 16u;
      unsigned lds = ldsbase + off;
      asm volatile("global_load_async_to_lds_b128 %0, %1, %2"
                   :: "v"(lds), "v"(off), "s"(gbase) : "memory");
    }
  }
  asm volatile("s_wait_asynccnt 0" ::: "memory");
  __syncthreads();

  const float scale = __uint_as_float(maxbits[0]) * (1.0f / 127.0f);
  const float inv_ok = scale;  // (kept: scale > 0 for this data)

  float rn[KK], re[KK], prio[KK];
  unsigned upc = 0, dnc = 0;
  float esum = 0.f;
  #pragma unroll
  for (int j = 0; j < KK; ++j) {
    float xv = tile[t * KK + j];
    float qq = fminf(fmaxf(xv / inv_ok, -127.f), 127.f);
    float r = rintf(qq);                 // round-half-even == jnp.round
    float e = r - qq;
    rn[j] = r; re[j] = e;
    if (e < 0.f && qq <  127.f) upc |= 1u << j;   // up candidate
    if (e > 0.f && qq > -127.f) dnc |= 1u << j;   // down candidate
    esum += e;
  }
  int nflip = (int)rintf(fabsf(esum));
  bool flip_up = (esum < 0.f);
  #pragma unroll
  for (int j = 0; j < KK; ++j) {
    bool cand = flip_up ? ((upc >> j) & 1) : ((dnc >> j) & 1);
    prio[j] = cand ? fabsf(re[j]) : 0.f;
  }
  // iterative top-n selection, ties -> lowest index (matches stable argsort)
  int last = -1;
  float b_val = 0.f;
  const float d = flip_up ? 1.f : -1.f;
  for (int f = 0; f < nflip && f < KK; ++f) {
    int best = -1; float bp = 0.f;
    #pragma unroll
    for (int j = 0; j < KK; ++j) {
      if (prio[j] > bp) { bp = prio[j]; best = j; }
    }
    if (best < 0) break;                 // safety; analytically unreachable
    #pragma unroll
    for (int j = 0; j < KK; ++j) {
      if (j == best) { rn[j] += d; re[j] += d; prio[j] = 0.f; b_val = fabsf(re[j]); }
    }
    last = best;
  }
  float er = 0.f;
  #pragma unroll
  for (int j = 0; j < KK; ++j) er += re[j];

  const int row = blockIdx.x * 256 + t;
  e_row[row] = er;
  if (last >= 0) {
    b_prio[row] = b_val;
    // boundary carries the *opposite* direction priority: dirUp=1 iff stage-1 flipped down
    b_meta[row] = last | ((flip_up ? 0 : 1) << 8);
  } else {
    b_prio[row] = 0.f;
    b_meta[row] = -1;
  }

  // scaled result -> LDS, then async store back out (same 16B chunking)
  #pragma unroll
  for (int j = 0; j < KK; ++j) tile[t * KK + j] = rn[j] * scale;
  __syncthreads();
  const uint64_t obase =
      (uint64_t)(uintptr_t)out + (uint64_t)blockIdx.x * blockBytes;
  #pragma unroll
  for (int i = 0; i < 3; ++i) {
    int idx = t + 256 * i;
    if (idx < 576) {
      unsigned off = (unsigned)idx * 16u;
      unsigned lds = ldsbase + off;
      asm volatile("global_store_async_from_lds_b128 %0, %1, %2"
                   :: "v"(off), "v"(lds), "s"(obase) : "memory");
    }
  }
  asm volatile("s_wait_asynccnt 0" ::: "memory");
}

// ---------------------------------------------------------------- stage 2 (squant-c)
// One block of 256 threads per OC channel (1024 rows x 9 elems = 9216 elements).
__global__ __launch_bounds__(256) void k_stage2(const float* __restrict__ x,
                                                float* __restrict__ out,
                                                const unsigned* __restrict__ maxbits,
                                                const float* __restrict__ e_row,
                                                const float* __restrict__ b_prio,
                                                const int* __restrict__ b_meta) {
  __shared__ float se[IC];                       // per-row residual sums
  __shared__ unsigned long long skey[IC];        // packed (prio_bits<<32)|~elem_idx
  __shared__ int belem[IC];                      // eligible boundary elem idx, else -1
  __shared__ int spre[IC + 1];                   // prefix of eligibility
  __shared__ float s_esum;
  __shared__ int s_m2;

  const int t = threadIdx.x;
  const int oc = blockIdx.x;
  const float scale = __uint_as_float(maxbits[0]) * (1.0f / 127.0f);

  for (int r = t; r < IC; r += 256) se[r] = e_row[oc * IC + r];
  if (t == 0) s_m2 = 0;
  __syncthreads();

  // e_sum2 = sum of 1024 fp32 values on the matrix pipe:
  // D = ones(16x4) x B + C  => D[m,n] accumulates column sums of B chunks.
  // Every wave computes it redundantly (EXEC all-ones per WMMA restriction).
  {
    const int lane = t & 31;
    v8f acc = {};
    v2f a; a.x = 1.f; a.y = 1.f;
    for (int c = 0; c < 16; ++c) {
      v2f b; b.x = se[c * 64 + lane]; b.y = se[c * 64 + 32 + lane];
      acc = __builtin_amdgcn_wmma_f32_16x16x4_f32(
          false, a, false, b, (short)0, acc, false, false);
    }
    // acc[0]: lanes 0-15 = row M=0 colsums, lanes 16-31 = row M=8 (identical)
    float tot = waveReduceSum(acc[0]) * 0.5f;
    if (t == 0) s_esum = tot;
  }
  __syncthreads();

  const float esum2 = s_esum;
  const int nflip2 = (int)rintf(fabsf(esum2));
  const bool flip_up2 = (esum2 < 0.f);

  // build selection keys for eligible boundaries (direction must match)
  int myelig = 0;
  for (int r = t; r < IC; r += 256) {
    int meta = b_meta[oc * IC + r];
    unsigned long long key = 0ull;
    int be = -1;
    if (meta >= 0 && (((meta >> 8) & 1) == (flip_up2 ? 1 : 0))) {
      int eidx = r * KK + (meta & 0xFF);
      unsigned pb = __float_as_uint(b_prio[oc * IC + r]);   // in (0.5,1] => pb != 0
      key = ((unsigned long long)pb << 32) |
            (unsigned long long)(0xFFFFFFFFu - (unsigned)eidx);
      be = eidx;
      ++myelig;
    }
    skey[r] = key;
    belem[r] = be;
  }
  atomicAdd(&s_m2, myelig);
  __syncthreads();
  const int m2 = s_m2;

  if (nflip2 > 0) {
    // exact top-n: rank = #keys strictly greater (keys are unique)
    for (int r = t; r < IC; r += 256) {
      unsigned long long mykey = skey[r];
      if (mykey != 0ull) {
        int rank = 0;
        for (int s = 0; s < IC; ++s) rank += (skey[s] > mykey) ? 1 : 0;
        if (rank < nflip2) {
          int eidx = belem[r];
          int g = oc * (IC * KK) + eidx;
          float xv = x[g];
          float qq = fminf(fmaxf(xv / scale, -127.f), 127.f);
          float rr = rintf(qq);
          float e = rr - qq;
          float nv = flip_up2 ? ((e < 0.f && qq <  127.f) ? rr + 1.f : rr)
                              : ((e > 0.f && qq > -127.f) ? rr - 1.f : rr);
          out[g] = nv * scale;
        }
      }
    }
    const int extra = nflip2 - m2;   // block-uniform
    if (extra > 0) {
      // rare: flip zero-priority elements in index order (skip eligible boundaries)
      if (t == 0) {
        int acc2 = 0;
        for (int r = 0; r < IC; ++r) { spre[r] = acc2; acc2 += (skey[r] != 0ull); }
        spre[IC] = acc2;
      }
      __syncthreads();
      for (int idx = t; idx < IC * KK; idx += 256) {
        int ic = idx / KK;
        int be = belem[ic];
        if (be == idx) continue;                        // eligible boundary: not zero-prio
        int zr = idx - spre[ic] - ((be >= 0 && be < idx) ? 1 : 0);
        if (zr < extra) {
          int g = oc * (IC * KK) + idx;
          float xv = x[g];
          float qq = fminf(fmaxf(xv / scale, -127.f), 127.f);
          float rr = rintf(qq);
          float e = rr - qq;
          float nv = flip_up2 ? ((e < 0.f && qq <  127.f) ? rr + 1.f : rr)
                              : ((e > 0.f && qq > -127.f) ? rr - 1.f : rr);
          out[g] = nv * scale;
        }
      }
    }
  }
}

// ---------------------------------------------------------------- launcher
extern "C" void kernel_launch(void* const* d_in, const int* in_sizes, int n_in,
                              void* d_out, int out_size, void* d_ws, size_t ws_size,
                              hipStream_t stream) {
  const float* x = (const float*)d_in[0];
  float* out = (float*)d_out;

  // workspace layout: [0..3] max|x| bits, then e_row / b_prio / b_meta (4MB each)
  char* wsb = (char*)d_ws;
  unsigned* maxbits = (unsigned*)wsb;
  float* e_row  = (float*)(wsb + 64);
  float* b_prio = (float*)(wsb + 64 + (size_t)NROWS * 4);
  int*   b_meta = (int*)  (wsb + 64 + (size_t)NROWS * 8);

  k_init  <<<1,        1,   0, stream>>>(maxbits);
  k_maxabs<<<1024,     256, 0, stream>>>(x, maxbits);
  k_stage1<<<NROWS/256,256, 0, stream>>>(x, out, maxbits, e_row, b_prio, b_meta);
  k_stage2<<<OC,       256, 0, stream>>>(x, out, maxbits, e_row, b_prio, b_meta);
}